// yololoss_16286515986956
// MI455X (gfx1250) — compile-verified
//
#include <hip/hip_runtime.h>
#include <hip/hip_bf16.h>

// YOLO-v1 loss for MI455X (gfx1250).
// Memory-bound streaming reduction: 96.4 MB in, 4 B out  (~4.1 us at 23.3 TB/s).
// Pass 1: per-block coalesced NT b128 loads -> LDS -> per-cell loss -> WMMA block reduce.
// Pass 2: deterministic final reduce of block sums (also via WMMA) -> d_out.

typedef float v2f __attribute__((ext_vector_type(2)));
typedef float v4f __attribute__((ext_vector_type(4)));
typedef float v8f __attribute__((ext_vector_type(8)));

#define GRID_S 28
#define CELL_LEN 30          // 5*B + C = 30 floats per cell
#define BLOCK 256            // 8 wave32 waves
#define FLOATS_PER_BLOCK (BLOCK * CELL_LEN)   // 7680 floats
#define V4_PER_BLOCK (FLOATS_PER_BLOCK / 4)   // 1920 float4

// Exact f32 sum of s[0..255] using V_WMMA_F32_16X16X4_F32 with A = ones(16x4):
//   D[m][n] = sum_k B[k][n] + C[m][n]  -> 4 chained WMMAs consume 4x64 values.
// With all-ones A the total is independent of the B operand lane layout (any
// bijection of 64 values into B slots sums identically), and every row of D
// equals the column-sum vector, so sum(all 256 D entries) == 16 * total.
// Must be called by a full wave (EXEC all ones).
__device__ __forceinline__ float wmma_reduce256(const float* __restrict__ s) {
  const int lane = threadIdx.x & 31;
  v2f ones;
  ones[0] = 1.0f;
  ones[1] = 1.0f;
  v8f acc = {};
#pragma unroll
  for (int g = 0; g < 4; ++g) {
    v2f b;
    b[0] = s[g * 64 + lane];
    b[1] = s[g * 64 + 32 + lane];
    acc = __builtin_amdgcn_wmma_f32_16x16x4_f32(
        /*neg_a=*/false, ones, /*neg_b=*/false, b,
        /*c_mod=*/(short)0, acc, /*reuse_a=*/false, /*reuse_b=*/false);
  }
  float tot = acc[0] + acc[1] + acc[2] + acc[3] + acc[4] + acc[5] + acc[6] + acc[7];
#pragma unroll
  for (int m = 16; m >= 1; m >>= 1) tot += __shfl_xor(tot, m, 32);
  return tot * (1.0f / 16.0f);
}

__global__ __launch_bounds__(BLOCK) void yolo_block_kernel(
    const float* __restrict__ pred, const float* __restrict__ targ,
    float* __restrict__ blockSums, int numCells) {
  __shared__ v4f sP4[V4_PER_BLOCK];   // 30 KB
  __shared__ v4f sT4[V4_PER_BLOCK];   // 30 KB

  const int tid = threadIdx.x;
  const long long base4 = (long long)blockIdx.x * V4_PER_BLOCK;
  const long long tot4 = ((long long)numCells * CELL_LEN) >> 2;
  const v4f* __restrict__ p4 = (const v4f*)pred;
  const v4f* __restrict__ t4 = (const v4f*)targ;

  // Fully coalesced, non-temporal (stream-once) 128-bit loads into LDS.
  for (int i = tid; i < V4_PER_BLOCK; i += BLOCK) {
    long long g = base4 + i;
    if (g < tot4) {
      sP4[i] = __builtin_nontemporal_load(&p4[g]);
      sT4[i] = __builtin_nontemporal_load(&t4[g]);
    }
  }
  __syncthreads();

  float partial = 0.0f;
  const int cell = blockIdx.x * BLOCK + tid;
  if (cell < numCells) {
    const float* __restrict__ p = (const float*)sP4 + tid * CELL_LEN;
    const float* __restrict__ t = (const float*)sT4 + tid * CELL_LEN;
    const float invS = 1.0f / (float)GRID_S;

    const float coo = (t[4] > 0.0f) ? 1.0f : 0.0f;

    // Target box 0 corners (match reference arithmetic order).
    const float tltx = t[0] * invS - 0.5f * t[2];
    const float tlty = t[1] * invS - 0.5f * t[3];
    const float trbx = t[0] * invS + 0.5f * t[2];
    const float trby = t[1] * invS + 0.5f * t[3];
    const float area2 = (trbx - tltx) * (trby - tlty);

    float iou[2];
#pragma unroll
    for (int b = 0; b < 2; ++b) {
      const float* pb = p + 5 * b;
      const float pltx = pb[0] * invS - 0.5f * pb[2];
      const float plty = pb[1] * invS - 0.5f * pb[3];
      const float prbx = pb[0] * invS + 0.5f * pb[2];
      const float prby = pb[1] * invS + 0.5f * pb[3];
      float ix = fminf(prbx, trbx) - fmaxf(pltx, tltx);
      float iy = fminf(prby, trby) - fmaxf(plty, tlty);
      ix = fmaxf(ix, 0.0f);
      iy = fmaxf(iy, 0.0f);
      const float inter = ix * iy;
      const float area1 = (prbx - pltx) * (prby - plty);
      iou[b] = inter / (area1 + area2 - inter);
    }

    // argmax picks first occurrence of the max (ties -> box 0).
    const int r = (iou[1] > iou[0]) ? 1 : 0;
    const float max_iou = fmaxf(iou[0], iou[1]);

    const float* pr = p + 5 * r;
    const float* tr = t + 5 * r;
    const float dx = pr[0] - tr[0];
    const float dy = pr[1] - tr[1];
    const float dw = sqrtf(pr[2]) - sqrtf(tr[2]);
    const float dh = sqrtf(pr[3]) - sqrtf(tr[3]);
    const float loc = dx * dx + dy * dy + dw * dw + dh * dh;

    const float cdiff = pr[4] - max_iou;
    const float contain = cdiff * cdiff;

    const float pcn = p[5 * (1 - r) + 4];
    const float notresp = pcn * pcn;

    const float n0 = p[4] - t[4];
    const float n1 = p[9] - t[9];
    const float noobj = n0 * n0 + n1 * n1;

    float cls = 0.0f;
#pragma unroll
    for (int k = 10; k < 30; ++k) {
      const float d = p[k] - t[k];
      cls += d * d;
    }

    partial = coo * (5.0f * loc + contain + cls) +
              0.5f * (coo * notresp + (1.0f - coo) * noobj);
  }

  // Block reduction: 256 partials -> 1, via 4 chained f32 WMMAs in wave 0.
  __syncthreads();                 // done reading staged data
  float* red = (float*)sP4;        // reuse LDS
  red[tid] = partial;
  __syncthreads();
  if (tid < 32) {                  // wave-uniform branch: EXEC all ones in wave 0
    const float bsum = wmma_reduce256(red);
    if (tid == 0) blockSums[blockIdx.x] = bsum;
  }
}

__global__ __launch_bounds__(BLOCK) void yolo_final_kernel(
    const float* __restrict__ blockSums, float* __restrict__ out,
    int nBlocks, float invN) {
  __shared__ float red[BLOCK];
  const int tid = threadIdx.x;
  float s = 0.0f;
  for (int i = tid; i < nBlocks; i += BLOCK) s += blockSums[i];  // fixed order
  red[tid] = s;
  __syncthreads();
  if (tid < 32) {
    const float tot = wmma_reduce256(red);
    if (tid == 0) out[0] = tot * invN;   // overwrite (no pre-zero needed)
  }
}

extern "C" void kernel_launch(void* const* d_in, const int* in_sizes, int n_in,
                              void* d_out, int out_size, void* d_ws, size_t ws_size,
                              hipStream_t stream) {
  (void)n_in; (void)out_size; (void)ws_size;
  const float* pred = (const float*)d_in[0];
  const float* targ = (const float*)d_in[1];

  const int totalFloats = in_sizes[0];                 // 512*28*28*30
  const int numCells = totalFloats / CELL_LEN;         // 401408
  const int N = numCells / (GRID_S * GRID_S);          // batch size = 512
  const int nBlocks = (numCells + BLOCK - 1) / BLOCK;  // 1568

  float* blockSums = (float*)d_ws;                     // nBlocks floats of scratch

  yolo_block_kernel<<<nBlocks, BLOCK, 0, stream>>>(pred, targ, blockSums, numCells);
  yolo_final_kernel<<<1, BLOCK, 0, stream>>>(blockSums, (float*)d_out, nBlocks,
                                             1.0f / (float)N);
}